// CMoSModel_40707700032360
// MI455X (gfx1250) — compile-verified
//
#include <hip/hip_runtime.h>
#include <math.h>

typedef float v2f __attribute__((ext_vector_type(2)));
typedef float v8f __attribute__((ext_vector_type(8)));

#define BS      256
#define SEQ     512
#define PRED    720
#define CIN     64
#define SEG     16
#define NMAP    8
#define KSZ     16
#define CSTRIDE 8
#define CONVD   63
#define NIN     32
#define NOUT    45
#define CPB     8      // channels per block (one per wave)
#define XPAD    520    // padded per-channel LDS stride (conflict-avoiding)
#define OPAD    9      // padded output stage stride

__global__ __launch_bounds__(256) void cmos_kernel(
    const float* __restrict__ x,      // [BS, SEQ, CIN]
    const float* __restrict__ conv_w, // [CIN, 1, KSZ]
    const float* __restrict__ conv_b, // [CIN]
    const float* __restrict__ gate_w, // [NMAP, CONVD]
    const float* __restrict__ gate_b, // [NMAP]
    const float* __restrict__ map_w,  // [NMAP, NOUT, NIN]
    const float* __restrict__ map_b,  // [NMAP, NOUT]
    float* __restrict__ out)          // [BS, PRED, CIN]
{
    __shared__ float xs[CPB * XPAD];        // normalized x, per channel
    __shared__ float convs[CPB * 64];       // conv taps, per channel
    __shared__ float outb[PRED * OPAD];     // staged outputs [p][ch]

    const int b    = blockIdx.x / (CIN / CPB);
    const int cg   = blockIdx.x % (CIN / CPB);
    const int tid  = threadIdx.x;
    const int w    = tid >> 5;              // wave id == local channel
    const int lane = tid & 31;
    const int c    = cg * CPB + w;          // global channel

    // ---- async global->LDS staging with per-lane transpose ([l][c] -> [c][l]) ----
    // GLOBAL_LOAD_ASYNC_TO_LDS_B32: per-lane LDS dest VGPR, tracked by ASYNCcnt.
    for (int i = tid; i < SEQ * CPB; i += 256) {
        int cl = i & (CPB - 1);
        int l  = i >> 3;
        const float* gp = &x[(b * SEQ + l) * CIN + cg * CPB + cl];
        unsigned lds_addr = (unsigned)(unsigned long long)(&xs[cl * XPAD + l]);
        asm volatile("global_load_async_to_lds_b32 %0, %1, off"
                     :: "v"(lds_addr), "v"(gp) : "memory");
    }
    asm volatile("s_wait_asynccnt 0" ::: "memory");
    __syncthreads();

    float* xc = &xs[w * XPAD];

    // ---- per-channel mean / var (wave32 reduction) ----
    float s = 0.f, s2 = 0.f;
    #pragma unroll
    for (int i = 0; i < SEQ / 32; ++i) {
        float v = xc[lane + i * 32];
        s += v; s2 += v * v;
    }
    #pragma unroll
    for (int off = 16; off > 0; off >>= 1) {
        s  += __shfl_xor(s,  off, 32);
        s2 += __shfl_xor(s2, off, 32);
    }
    const float mean  = s  * (1.f / SEQ);
    const float var   = fmaxf(s2 * (1.f / SEQ) - mean * mean, 0.f);
    const float stdev = sqrtf(var + 1e-10f);
    const float rstd  = 1.f / stdev;

    // ---- normalize in place ----
    #pragma unroll
    for (int i = 0; i < SEQ / 32; ++i) {
        int l = lane + i * 32;
        xc[l] = (xc[l] - mean) * rstd;
    }

    // ---- depthwise conv: 63 taps, 2 per lane ----
    float cw[KSZ];
    #pragma unroll
    for (int k = 0; k < KSZ; ++k) cw[k] = conv_w[c * KSZ + k];
    const float cb = conv_b[c];
    float c0 = 0.f, c1 = 0.f;
    {
        const int d0 = lane, d1 = lane + 32;
        #pragma unroll
        for (int k = 0; k < KSZ; ++k) {
            c0 += xc[d0 * CSTRIDE + k] * cw[k];
            if (d1 < CONVD) c1 += xc[d1 * CSTRIDE + k] * cw[k];
        }
    }
    convs[w * 64 + lane]      = c0 + cb;                          // d = 0..31
    convs[w * 64 + lane + 32] = (lane + 32 < CONVD) ? (c1 + cb) : 0.f;

    // ---- gate logits + softmax + top-2 (uniform across lanes; keeps EXEC full) ----
    float logit[NMAP];
    #pragma unroll
    for (int m = 0; m < NMAP; ++m) {
        float acc = gate_b[m];
        for (int d = 0; d < CONVD; ++d)
            acc += convs[w * 64 + d] * gate_w[m * CONVD + d];
        logit[m] = acc;
    }
    float mx = logit[0];
    #pragma unroll
    for (int m = 1; m < NMAP; ++m) mx = fmaxf(mx, logit[m]);
    float p[NMAP], Z = 0.f;
    #pragma unroll
    for (int m = 0; m < NMAP; ++m) { p[m] = __expf(logit[m] - mx); Z += p[m]; }
    const float rZ = 1.f / Z;
    #pragma unroll
    for (int m = 0; m < NMAP; ++m) p[m] *= rZ;
    int m0 = 0; float v0 = p[0];
    #pragma unroll
    for (int m = 1; m < NMAP; ++m) if (p[m] > v0) { v0 = p[m]; m0 = m; }
    int m1 = 0; float v1 = -1.f;
    #pragma unroll
    for (int m = 0; m < NMAP; ++m) if (m != m0 && p[m] > v1) { v1 = p[m]; m1 = m; }
    const float e0 = __expf(v0), e1 = __expf(v1);
    const float g0 = e0 / (e0 + e1), g1 = e1 / (e0 + e1);

    // ---- fp32 WMMA: D[s=16, o=16] = A[s,K=32] x B[K,o] in K=4 chunks, 3 o-tiles ----
    // A layout (16x4 f32): lanes 0-15 M=0..15 K={0,1}; lanes 16-31 M=0..15 K={2,3}
    // B/C/D: N = lane%16; B vgpr v: K=v (+2 upper lanes); C/D vgpr v: M=v (+8 upper lanes)
    const int srow  = lane & 15;          // s for A rows; o column for B/C/D
    const int khalf = (lane >> 4) * 2;    // K offset 0 or 2
    #pragma unroll
    for (int tile = 0; tile < 3; ++tile) {
        const int o  = tile * 16 + srow;
        const bool ov = (o < NOUT);
        const float beff = ov ? (g0 * map_b[m0 * NOUT + o] + g1 * map_b[m1 * NOUT + o]) : 0.f;
        v8f acc;
        #pragma unroll
        for (int v = 0; v < 8; ++v) acc[v] = beff;   // bias broadcast over s rows

        #pragma unroll
        for (int kk = 0; kk < 8; ++kk) {
            const int n0 = kk * 4 + khalf;
            v2f a, bb;
            a.x = xc[n0 * SEG + srow];               // x_[s, n0]   (x_[s,n] = xn[n*SEG+s])
            a.y = xc[(n0 + 1) * SEG + srow];         // x_[s, n0+1]
            if (ov) {
                const float* w0 = &map_w[(m0 * NOUT + o) * NIN + n0];
                const float* w1 = &map_w[(m1 * NOUT + o) * NIN + n0];
                bb.x = g0 * w0[0] + g1 * w1[0];      // W_eff[o, n0]
                bb.y = g0 * w0[1] + g1 * w1[1];      // W_eff[o, n0+1]
            } else { bb.x = 0.f; bb.y = 0.f; }
            acc = __builtin_amdgcn_wmma_f32_16x16x4_f32(
                false, a, false, bb, (short)0, acc, false, false);
        }

        // epilogue: de-normalize and stage to LDS (p = o*SEG + s)
        #pragma unroll
        for (int v = 0; v < 8; ++v) {
            const int sr = (lane < 16) ? v : v + 8;
            if (ov) outb[(o * SEG + sr) * OPAD + w] = acc[v] * stdev + mean;
        }
    }
    __syncthreads();

    // ---- cooperative store, widened to b128: out[b, p, cg*8 + q*4 .. +3] ----
    for (int i = tid; i < PRED * (CPB / 4); i += 256) {   // 1440 float4 stores
        int q  = i & 1;
        int pp = i >> 1;
        float4 vv;
        vv.x = outb[pp * OPAD + q * 4 + 0];
        vv.y = outb[pp * OPAD + q * 4 + 1];
        vv.z = outb[pp * OPAD + q * 4 + 2];
        vv.w = outb[pp * OPAD + q * 4 + 3];
        *reinterpret_cast<float4*>(&out[(b * PRED + pp) * CIN + cg * CPB + q * 4]) = vv;
    }
}

extern "C" void kernel_launch(void* const* d_in, const int* in_sizes, int n_in,
                              void* d_out, int out_size, void* d_ws, size_t ws_size,
                              hipStream_t stream) {
    const float* x      = (const float*)d_in[0];
    const float* conv_w = (const float*)d_in[1];
    const float* conv_b = (const float*)d_in[2];
    const float* gate_w = (const float*)d_in[3];
    const float* gate_b = (const float*)d_in[4];
    const float* map_w  = (const float*)d_in[5];
    const float* map_b  = (const float*)d_in[6];
    float* out = (float*)d_out;

    dim3 grid(BS * (CIN / CPB));   // 2048 blocks: (b, channel-group)
    dim3 block(256);               // 8 wave32 waves, one channel each
    cmos_kernel<<<grid, block, 0, stream>>>(x, conv_w, conv_b, gate_w, gate_b,
                                            map_w, map_b, out);
}